// ComputeLoss_50302656970893
// MI455X (gfx1250) — compile-verified
//
#include <hip/hip_runtime.h>
#include <math.h>

// ---------------------------------------------------------------------------
// YOLOv5 ComputeLoss for MI455X (gfx1250, wave32).
// Traffic-minimal: reads only objectness planes (3/255 of each tensor) plus
// K=15360 gathered cells/level. WMMA f32_16x16x4 (all-ones B) used as the
// block-sum engine: D[m][n] = sum_k A[m][k]; 4 chained WMMAs reduce 256
// f32 partials per block with exact f32 accumulation, deterministic order.
// ---------------------------------------------------------------------------

typedef float v2f __attribute__((ext_vector_type(2)));
typedef float v8f __attribute__((ext_vector_type(8)));

__device__ __forceinline__ float softplusf(float x) {
    // numerically stable: log(1+e^x) = log1p(e^{-|x|}) + max(x,0)
    return log1pf(expf(-fabsf(x))) + fmaxf(x, 0.0f);
}
__device__ __forceinline__ float sigmoidf(float x) {
    return 1.0f / (1.0f + expf(-x));
}

// Deterministic 256-thread block sum using V_WMMA_F32_16X16X4_F32.
// Tile t (64 values) laid out as A[m][k] = sdata[t*64 + k*16 + m].
// A layout (ISA 7.12.2): lanes 0-15 hold (k=0,k=1) of row m=lane in 2 VGPRs,
// lanes 16-31 hold (k=2,k=3) of row m=lane-16. B = all ones -> row sums.
// D layout: VGPR j = row j (lanes 0-15) / row 8+j (lanes 16-31).
__device__ __forceinline__ float block_reduce_wmma_256(float val) {
    __shared__ float sdata[258];
    int tid = threadIdx.x;
    sdata[tid] = val;
    __syncthreads();
    if (tid < 32) {                      // wave 0: EXEC all ones
        int base = ((tid & 16) << 1) + (tid & 15);  // lanes>=16 -> +32 (k=2,3)
        v2f b; b.x = 1.0f; b.y = 1.0f;   // ones B (layout-independent)
        v8f c = {};
#pragma unroll
        for (int t = 0; t < 4; ++t) {
            v2f a;
            a.x = sdata[t * 64 + base];
            a.y = sdata[t * 64 + base + 16];
            c = __builtin_amdgcn_wmma_f32_16x16x4_f32(
                    false, a, false, b, (short)0, c, false, false);
        }
        float s = c.s0 + c.s1 + c.s2 + c.s3 + c.s4 + c.s5 + c.s6 + c.s7;
        if (tid == 0)  sdata[256] = s;   // rows 0-7
        if (tid == 16) sdata[257] = s;   // rows 8-15
    }
    __syncthreads();
    float total = sdata[256] + sdata[257];
    __syncthreads();                     // safe reuse across calls
    return total;
}

__device__ __forceinline__ float giou_fn(float b1x, float b1y, float b1w, float b1h,
                                         float b2x, float b2y, float b2w, float b2h) {
    const float eps = 1e-7f;
    float b1x1 = b1x - b1w * 0.5f, b1x2 = b1x + b1w * 0.5f;
    float b1y1 = b1y - b1h * 0.5f, b1y2 = b1y + b1h * 0.5f;
    float b2x1 = b2x - b2w * 0.5f, b2x2 = b2x + b2w * 0.5f;
    float b2y1 = b2y - b2h * 0.5f, b2y2 = b2y + b2h * 0.5f;
    float iw = fmaxf(fminf(b1x2, b2x2) - fmaxf(b1x1, b2x1), 0.0f);
    float ih = fmaxf(fminf(b1y2, b2y2) - fmaxf(b1y1, b2y1), 0.0f);
    float inter = iw * ih;
    float uni = b1w * b1h + b2w * b2h - inter + eps;
    float iou = inter / uni;
    float cw = fmaxf(b1x2, b2x2) - fminf(b1x1, b2x1);
    float ch = fmaxf(b1y2, b2y2) - fminf(b1y1, b2y1);
    float ca = cw * ch + eps;
    return iou - (ca - uni) / ca;
}

// One thread per candidate k in [0, 5*3*1024). Computes box+cls partials and
// scatter-max objectness targets. K = 15360 = 60 blocks * 256.
__global__ void __launch_bounds__(256)
assign_kernel(const float* __restrict__ p, const float* __restrict__ targets,
              float* __restrict__ obj_gt, float* __restrict__ partials,
              int W, int H,
              float a0x, float a0y, float a1x, float a1y, float a2x, float a2y) {
    const int A = 3, M = 1024, Nimg = 32;
    int k = blockIdx.x * blockDim.x + threadIdx.x;
    int j = k / (A * M);
    int rest = k - j * (A * M);
    int a = rest / M;
    int m = rest - a * M;

    const float* t = targets + m * 6;
    float img_f = t[0];
    int   cls   = (int)t[1];
    float gx_f = t[2] * (float)W, gy_f = t[3] * (float)H;
    float gw   = t[4] * (float)W, gh   = t[5] * (float)H;

    float ax = (a == 0) ? a0x : ((a == 1) ? a1x : a2x);
    float ay = (a == 0) ? a0y : ((a == 1) ? a1y : a2y);

    float rw = ax / gw, rh = ay / gh;
    float rmax = fmaxf(fmaxf(rw, 1.0f / rw), fmaxf(rh, 1.0f / rh));
    bool bm = rmax < 4.0f;   // ANCHOR_T

    float remx = gx_f - floorf(gx_f);
    float remy = gy_f - floorf(gy_f);
    float ox = 0.0f, oy = 0.0f;
    bool side = true;
    if (j == 1)      { side = (remx < 0.5f) && (gx_f > 1.0f);            ox =  0.5f; }
    else if (j == 2) { side = (remy < 0.5f) && (gy_f > 1.0f);            oy =  0.5f; }
    else if (j == 3) { side = (remx > 0.5f) && (gx_f < (float)W - 1.0f); ox = -0.5f; }
    else if (j == 4) { side = (remy > 0.5f) && (gy_f < (float)H - 1.0f); oy = -0.5f; }
    bool cm = bm && side;

    float fgx = floorf(gx_f - ox), fgy = floorf(gy_f - oy);
    int gi  = min(max((int)fgx, 0), W - 1);
    int gj  = min(max((int)fgy, 0), H - 1);
    int img = min(max((int)img_f, 0), Nimg - 1);
    float tx = gx_f - fgx, ty = gy_f - fgy;   // offsets use UNclipped grid

    size_t hw = (size_t)H * (size_t)W;
    const float* pc = p + ((size_t)img * 255 + (size_t)a * 85) * hw
                        + (size_t)gj * (size_t)W + (size_t)gi;

    float l0 = pc[0], l1 = pc[hw], l2 = pc[2 * hw], l3 = pc[3 * hw];
    float px = sigmoidf(l0) * 2.0f - 0.5f;
    float py = sigmoidf(l1) * 2.0f - 0.5f;
    float sw = sigmoidf(l2) * 2.0f;  float pw = sw * sw * ax;
    float sh = sigmoidf(l3) * 2.0f;  float ph = sh * sh * ay;

    float giou = giou_fn(px, py, pw, ph, tx, ty, gw, gh);
    float fm = cm ? 1.0f : 0.0f;
    float lbox_c = fm * (1.0f - giou);

    // sum_c BCE(x_c, onehot) = sum_c softplus(x_c) - x_cls
    float clssum = 0.0f;
#pragma unroll 4
    for (int c = 0; c < 80; ++c) clssum += softplusf(pc[(size_t)(5 + c) * hw]);
    float lcls_c = fm * (clssum - pc[(size_t)(5 + cls) * hw]);

    // objectness target: scatter-max of clip(giou,0) — order-independent
    if (cm) {
        float ov = fmaxf(giou, 0.0f);
        if (ov > 0.0f) {
            size_t oidx = ((size_t)(img * 3 + a) * (size_t)H + (size_t)gj) * (size_t)W
                          + (size_t)gi;
            atomicMax((unsigned int*)(obj_gt + oidx), __float_as_uint(ov));
        }
    }

    float b0 = block_reduce_wmma_256(lbox_c);
    float b1 = block_reduce_wmma_256(lcls_c);
    float b2 = block_reduce_wmma_256(fm);
    if (threadIdx.x == 0) {
        partials[blockIdx.x * 3 + 0] = b0;
        partials[blockIdx.x * 3 + 1] = b1;
        partials[blockIdx.x * 3 + 2] = b2;
    }
}

// BCE(objectness) over N*A*H*W; reads only the channel-4 plane (coalesced).
__global__ void __launch_bounds__(256)
obj_kernel(const float* __restrict__ p, const float* __restrict__ obj_gt,
           float* __restrict__ partials, int W, int H) {
    size_t i = (size_t)blockIdx.x * blockDim.x + threadIdx.x;
    size_t hw = (size_t)H * (size_t)W;
    size_t ahw = 3 * hw;
    int n = (int)(i / ahw);
    size_t r = i - (size_t)n * ahw;
    int a = (int)(r / hw);
    size_t pos = r - (size_t)a * hw;

    float logit = p[((size_t)n * 255 + (size_t)a * 85 + 4) * hw + pos];
    float yt = obj_gt[i];
    float l = softplusf(logit) - yt * logit;   // == yt*sp(-x)+(1-yt)*sp(x)

    float s = block_reduce_wmma_256(l);
    if (threadIdx.x == 0) partials[blockIdx.x] = s;
}

// Single-block final combine: fixed-order partial sums -> scalar loss.
__global__ void __launch_bounds__(256)
combine_kernel(const float* __restrict__ aP0, const float* __restrict__ aP1,
               const float* __restrict__ aP2,
               const float* __restrict__ oP0, const float* __restrict__ oP1,
               const float* __restrict__ oP2,
               float* __restrict__ out) {
    int tid = threadIdx.x;
    const float* aP[3] = {aP0, aP1, aP2};
    const float* oP[3] = {oP0, oP1, oP2};
    const int   onb[3] = {2400, 600, 150};
    const float den[3] = {614400.0f, 153600.0f, 38400.0f};  // N*A*H*W
    const float bal[3] = {4.0f, 1.0f, 0.4f};

    float lbox = 0.0f, lcls = 0.0f, lobj = 0.0f;
    for (int l = 0; l < 3; ++l) {
        float v0 = 0.0f, v1 = 0.0f, v2 = 0.0f;
        for (int i = tid; i < 60; i += 256) {
            v0 += aP[l][i * 3 + 0];
            v1 += aP[l][i * 3 + 1];
            v2 += aP[l][i * 3 + 2];
        }
        float lbox_s = block_reduce_wmma_256(v0);
        float lcls_s = block_reduce_wmma_256(v1);
        float cnt_s  = block_reduce_wmma_256(v2);

        float vo = 0.0f;
        for (int i = tid; i < onb[l]; i += 256) vo += oP[l][i];
        float obj_s = block_reduce_wmma_256(vo);

        float cnt = fmaxf(cnt_s, 1.0f);
        lbox += lbox_s / cnt;
        lcls += lcls_s / (cnt * 80.0f);
        lobj += (obj_s / den[l]) * bal[l];
    }
    if (tid == 0)
        out[0] = (lbox * 0.05f + lcls * 0.5f + lobj * 1.0f) * 32.0f;
}

extern "C" void kernel_launch(void* const* d_in, const int* in_sizes, int n_in,
                              void* d_out, int out_size, void* d_ws, size_t ws_size,
                              hipStream_t stream) {
    const float* p0      = (const float*)d_in[0];  // (32,255,80,80)
    const float* p1      = (const float*)d_in[1];  // (32,255,40,40)
    const float* p2      = (const float*)d_in[2];  // (32,255,20,20)
    const float* targets = (const float*)d_in[3];  // (1024,6)
    float* out = (float*)d_out;

    float* ws  = (float*)d_ws;
    float* og0 = ws;                 // 614400 obj_gt level0
    float* og1 = og0 + 614400;       // 153600 obj_gt level1
    float* og2 = og1 + 153600;       // 38400  obj_gt level2
    float* aP0 = og2 + 38400;        // 60*3 assign partials per level
    float* aP1 = aP0 + 180;
    float* aP2 = aP1 + 180;
    float* oP0 = aP2 + 180;          // obj partials
    float* oP1 = oP0 + 2400;
    float* oP2 = oP1 + 600;

    // zero objectness targets (scatter-max accumulators) each launch
    hipMemsetAsync(ws, 0, (size_t)806400 * sizeof(float), stream);

    // anchors_px / stride (exact binary fractions)
    assign_kernel<<<60, 256, 0, stream>>>(p0, targets, og0, aP0, 80, 80,
        1.25f, 1.625f, 2.0f, 3.75f, 4.125f, 2.875f);
    assign_kernel<<<60, 256, 0, stream>>>(p1, targets, og1, aP1, 40, 40,
        1.875f, 3.8125f, 3.875f, 2.8125f, 3.6875f, 7.4375f);
    assign_kernel<<<60, 256, 0, stream>>>(p2, targets, og2, aP2, 20, 20,
        3.625f, 2.8125f, 4.875f, 6.1875f, 11.65625f, 10.1875f);

    obj_kernel<<<2400, 256, 0, stream>>>(p0, og0, oP0, 80, 80);
    obj_kernel<<<600,  256, 0, stream>>>(p1, og1, oP1, 40, 40);
    obj_kernel<<<150,  256, 0, stream>>>(p2, og2, oP2, 20, 20);

    combine_kernel<<<1, 256, 0, stream>>>(aP0, aP1, aP2, oP0, oP1, oP2, out);
}